// MySAGEConv_7473243095279
// MI455X (gfx1250) — compile-verified
//
#include <hip/hip_runtime.h>
#include <hip/hip_bf16.h>

#define N_NODES 100000
#define NUM_E   1600000
#define IN_C    128
#define OUT_C   128
#define A_STRIDE 260   // 16x256 tile, +4 pad -> lane stride 4 banks, conflict-free

typedef __attribute__((ext_vector_type(2))) float v2f;
typedef __attribute__((ext_vector_type(8))) float v8f;

// ---------------- zero workspace (agg [N,128] + cnt [N]) ----------------
__global__ void sage_zero_ws(float4* __restrict__ ws, int n4) {
    int i = blockIdx.x * blockDim.x + threadIdx.x;
    int stride = gridDim.x * blockDim.x;
    float4 z = make_float4(0.f, 0.f, 0.f, 0.f);
    for (; i < n4; i += stride) ws[i] = z;
}

// ---------------- edge scatter: one wave32 per edge ----------------
// lane l handles channels [4l, 4l+4): one b128 gather + 4 f32 atomics.
__global__ void __launch_bounds__(256)
sage_scatter(const float* __restrict__ x, const int* __restrict__ ei,
             float* __restrict__ agg, float* __restrict__ cnt) {
    int lane = threadIdx.x & 31;
    int gw = (blockIdx.x * blockDim.x + threadIdx.x) >> 5;
    int nw = (gridDim.x * blockDim.x) >> 5;
    for (int e = gw; e < NUM_E; e += nw) {
        int s = ei[e];           // src row of edge e
        int d = ei[NUM_E + e];   // dst row of edge e
        float4 v = ((const float4*)(x + (size_t)s * IN_C))[lane];
        float* a = agg + (size_t)d * IN_C + lane * 4;
        atomicAdd(a + 0, v.x);
        atomicAdd(a + 1, v.y);
        atomicAdd(a + 2, v.z);
        atomicAdd(a + 3, v.w);
        if (lane == 0) atomicAdd(cnt + d, 1.0f);
    }
}

// ---------------- fused mean + dual GEMM via V_WMMA_F32_16X16X4_F32 ----------------
// Block: 16 nodes (M-tile), 256 threads = 8 waves; wave w owns output cols [16w,16w+16).
// A = [agg_mean | x] (16x256) staged in LDS; B = rows of W_src / W_dst (out = feat @ W^T).
__global__ void __launch_bounds__(256)
sage_gemm(const float* __restrict__ x, const float* __restrict__ agg,
          const float* __restrict__ cnt,
          const float* __restrict__ Wsrc, const float* __restrict__ bsrc,
          const float* __restrict__ Wdst, const float* __restrict__ bdst,
          float* __restrict__ out) {
    __shared__ float sA[16 * A_STRIDE];
    __shared__ float sInv[16];

    int t = threadIdx.x;
    int base = blockIdx.x * 16;

    if (t < 16) {
        float c = cnt[base + t];
        sInv[t] = 1.0f / fmaxf(c, 1.0f);
    }
    __syncthreads();

    // Stage A tile: col = t (0..255), rows j = 0..15.
    {
        int col = t;
        for (int j = 0; j < 16; ++j) {
            float v;
            if (col < IN_C) v = agg[(size_t)(base + j) * IN_C + col] * sInv[j];
            else            v = x[(size_t)(base + j) * IN_C + (col - IN_C)];
            sA[j * A_STRIDE + col] = v;
        }
    }
    __syncthreads();

    int lane = t & 31;
    int wave = t >> 5;       // 0..7 -> output-channel tile
    int n0   = wave * 16;
    int half = lane >> 4;    // 0: K0/K1 lanes, 1: K2/K3 lanes (ISA 32-bit A layout)
    int mr   = lane & 15;    // A row / B column within tile

    v8f acc = {};
    const float* Wrow_s = Wsrc + (size_t)(n0 + mr) * IN_C;
    const float* Wrow_d = Wdst + (size_t)(n0 + mr) * IN_C;

    #pragma unroll 8
    for (int k0 = 0; k0 < IN_C; k0 += 4) {
        v2f a = *(const v2f*)&sA[mr * A_STRIDE + k0 + half * 2];
        v2f b = *(const v2f*)&Wrow_s[k0 + half * 2];
        acc = __builtin_amdgcn_wmma_f32_16x16x4_f32(
            false, a, false, b, (short)0, acc, false, false);
    }
    #pragma unroll 8
    for (int k0 = 0; k0 < IN_C; k0 += 4) {
        v2f a = *(const v2f*)&sA[mr * A_STRIDE + IN_C + k0 + half * 2];
        v2f b = *(const v2f*)&Wrow_d[k0 + half * 2];
        acc = __builtin_amdgcn_wmma_f32_16x16x4_f32(
            false, a, false, b, (short)0, acc, false, false);
    }

    // C/D layout: VGPR r, lanes<16 -> M=r, lanes>=16 -> M=r+8; N = lane%16.
    float bias = bsrc[n0 + mr] + bdst[n0 + mr];
    #pragma unroll
    for (int r = 0; r < 8; ++r) {
        int m = r + half * 8;
        out[(size_t)(base + m) * OUT_C + n0 + mr] = acc[r] + bias;
    }
}

extern "C" void kernel_launch(void* const* d_in, const int* in_sizes, int n_in,
                              void* d_out, int out_size, void* d_ws, size_t ws_size,
                              hipStream_t stream) {
    const float* x    = (const float*)d_in[0];
    const int*   ei   = (const int*)d_in[1];   // [2, E]
    const float* Wsrc = (const float*)d_in[2];
    const float* bsrc = (const float*)d_in[3];
    const float* Wdst = (const float*)d_in[4];
    const float* bdst = (const float*)d_in[5];
    float* out = (float*)d_out;

    float* agg = (float*)d_ws;                         // [N, 128]
    float* cnt = agg + (size_t)N_NODES * IN_C;         // [N]

    int n4 = (N_NODES * IN_C + N_NODES) / 4;           // 12.9M floats -> 3.225M float4
    sage_zero_ws<<<2048, 256, 0, stream>>>((float4*)d_ws, n4);
    sage_scatter<<<4096, 256, 0, stream>>>(x, ei, agg, cnt);
    sage_gemm<<<N_NODES / 16, 256, 0, stream>>>(x, agg, cnt, Wsrc, bsrc, Wdst, bdst, out);
}